// GCN_90357521973356
// MI455X (gfx1250) — compile-verified
//
#include <hip/hip_runtime.h>

typedef __attribute__((ext_vector_type(2))) float v2f;
typedef __attribute__((ext_vector_type(8))) float v8f;

#define TM 128
#define TN 64
#define TK 32
#define APAD 2  // row stride TK+2 = 34 floats: even (keeps 8B ds alignment), kills bank conflicts

__global__ __launch_bounds__(256) void gcn_zero_f32(float* __restrict__ p, int n) {
  int i = blockIdx.x * blockDim.x + threadIdx.x;
  if (i < n) p[i] = 0.0f;
}

__global__ __launch_bounds__(256) void gcn_degree(const int* __restrict__ src,
                                                  const int* __restrict__ dst,
                                                  float* __restrict__ deg_s,
                                                  float* __restrict__ deg_d, int E) {
  int i = blockIdx.x * blockDim.x + threadIdx.x;
  if (i < E) {
    atomicAdd(&deg_s[src[i]], 1.0f);
    atomicAdd(&deg_d[dst[i]], 1.0f);
  }
}

__global__ __launch_bounds__(256) void gcn_inv_sqrt(float* __restrict__ d, int n) {
  int i = blockIdx.x * blockDim.x + threadIdx.x;
  if (i < n) d[i] = rsqrtf(fmaxf(d[i], 1.0f));
}

// C[M,N] = (diag(rs) * A[M,K]) @ B[K,N], fp32, via V_WMMA_F32_16X16X4_F32.
// 8 waves per block; wave w computes rows [16w,16w+16) x 64 cols (4 v8f accumulators).
__global__ __launch_bounds__(256) void gcn_gemm_rowscale(
    const float* __restrict__ A, const float* __restrict__ rs,
    const float* __restrict__ B, float* __restrict__ C, int M, int K, int N) {
  __shared__ float As[TM][TK + APAD];
  __shared__ float Bs[TN][TK + APAD];  // transposed: Bs[n][k]

  const int tid = threadIdx.x;
  const int lane = tid & 31;
  const int wave = tid >> 5;
  const int row0 = blockIdx.x * TM;
  const int col0 = blockIdx.y * TN;
  const int lm = lane & 15;        // n index within 16x16 tile
  const int kg = (lane >> 4) * 2;  // k pair select: 0 or 2

  v8f acc0 = {}, acc1 = {}, acc2 = {}, acc3 = {};

  for (int kc = 0; kc < K; kc += TK) {
    // Stage A (TM x TK), scaled by rs[row]; each thread: 4 x float4
#pragma unroll
    for (int p = 0; p < 4; ++p) {
      int r = (tid >> 3) + 32 * p;
      int q = (tid & 7) * 4;
      int gr = row0 + r;
      float4 v = make_float4(0.f, 0.f, 0.f, 0.f);
      float s = 0.f;
      if (gr < M) {
        v = *(const float4*)(A + (size_t)gr * K + kc + q);
        s = rs[gr];
      }
      As[r][q + 0] = v.x * s;
      As[r][q + 1] = v.y * s;
      As[r][q + 2] = v.z * s;
      As[r][q + 3] = v.w * s;
    }
    // Stage B transposed (TK x TN -> Bs[n][k])
#pragma unroll
    for (int p = 0; p < 2; ++p) {
      int kr = (tid >> 4) + 16 * p;
      int nq = (tid & 15) * 4;
      float4 v = *(const float4*)(B + (size_t)(kc + kr) * N + col0 + nq);
      Bs[nq + 0][kr] = v.x;
      Bs[nq + 1][kr] = v.y;
      Bs[nq + 2][kr] = v.z;
      Bs[nq + 3][kr] = v.w;
    }
    __syncthreads();
#pragma unroll
    for (int ks = 0; ks < TK; ks += 4) {
      v2f a = *(const v2f*)&As[16 * wave + lm][ks + kg];
      v2f b0 = *(const v2f*)&Bs[0 + lm][ks + kg];
      v2f b1 = *(const v2f*)&Bs[16 + lm][ks + kg];
      v2f b2 = *(const v2f*)&Bs[32 + lm][ks + kg];
      v2f b3 = *(const v2f*)&Bs[48 + lm][ks + kg];
      acc0 = __builtin_amdgcn_wmma_f32_16x16x4_f32(false, a, false, b0, (short)0, acc0, false, false);
      acc1 = __builtin_amdgcn_wmma_f32_16x16x4_f32(false, a, false, b1, (short)0, acc1, false, false);
      acc2 = __builtin_amdgcn_wmma_f32_16x16x4_f32(false, a, false, b2, (short)0, acc2, false, false);
      acc3 = __builtin_amdgcn_wmma_f32_16x16x4_f32(false, a, false, b3, (short)0, acc3, false, false);
    }
    __syncthreads();
  }

  // Store D: lane l holds (m = r + 8*(l/16), n = l%16) in acc[r]
  const int mh = 8 * (lane >> 4);
  v8f accs[4] = {acc0, acc1, acc2, acc3};
#pragma unroll
  for (int t = 0; t < 4; ++t) {
#pragma unroll
    for (int r = 0; r < 8; ++r) {
      int gr = row0 + 16 * wave + mh + r;
      if (gr < M) C[(size_t)gr * N + col0 + 16 * t + lm] = accs[t][r];
    }
  }
}

// agg[sidx[e], :] += m[gidx[e], :]  (float4 per thread, f32 global atomics)
__global__ __launch_bounds__(256) void gcn_scatter_add(
    const float* __restrict__ m, const int* __restrict__ gidx,
    const int* __restrict__ sidx, float* __restrict__ agg, int E, int N) {
  int f4 = N >> 2;           // float4s per row (64 or 32)
  int epb = 256 / f4;        // edges per block (4 or 8)
  int e = blockIdx.x * epb + threadIdx.x / f4;
  if (e >= E) return;
  int c = (threadIdx.x % f4) * 4;
  int g = gidx[e];
  int s = sidx[e];
  float4 v = *(const float4*)(m + (size_t)g * N + c);
  float* out = agg + (size_t)s * N + c;
  atomicAdd(out + 0, v.x);
  atomicAdd(out + 1, v.y);
  atomicAdd(out + 2, v.z);
  atomicAdd(out + 3, v.w);
}

// out = relu(agg * inv[row] + bias[col])
__global__ __launch_bounds__(256) void gcn_epilogue(
    const float* __restrict__ agg, const float* __restrict__ inv,
    const float* __restrict__ bias, float* __restrict__ out, int n, int N) {
  int i = blockIdx.x * blockDim.x + threadIdx.x;
  if (i >= n * N) return;
  int r = i / N;
  int c = i - r * N;
  out[i] = fmaxf(fmaf(agg[i], inv[r], bias[c]), 0.0f);
}

static inline int cdiv(int a, int b) { return (a + b - 1) / b; }

extern "C" void kernel_launch(void* const* d_in, const int* in_sizes, int n_in,
                              void* d_out, int out_size, void* d_ws, size_t ws_size,
                              hipStream_t stream) {
  (void)n_in; (void)out_size; (void)ws_size;
  const float* h_s   = (const float*)d_in[0];
  const float* h_d   = (const float*)d_in[1];
  const float* W1_sd = (const float*)d_in[2];
  const float* b1_sd = (const float*)d_in[3];
  const float* W1_ds = (const float*)d_in[4];
  const float* b1_ds = (const float*)d_in[5];
  const float* W3_sd = (const float*)d_in[6];
  const float* b3_sd = (const float*)d_in[7];
  const float* W3_ds = (const float*)d_in[8];
  const float* b3_ds = (const float*)d_in[9];
  const int* src = (const int*)d_in[10];
  const int* dst = (const int*)d_in[11];

  const int CD = 256, HD = 256, OD = 128;
  const int NS = in_sizes[0] / CD;   // 50000
  const int ND = in_sizes[1] / CD;   // 50000
  const int E  = in_sizes[10];       // 300000

  // Workspace layout (floats): [inv_s | inv_d | buf0(NS*HD) | h1_d(ND*HD) | h1_s(NS*HD)]
  float* ws    = (float*)d_ws;
  float* inv_s = ws;                              // NS (holds deg then inv)
  float* inv_d = ws + NS;                         // ND
  float* buf0  = ws + ((NS + ND + 255) & ~255);   // message buffer, NS*HD floats
  float* h1_d  = buf0 + (size_t)NS * HD;          // agg then h1_d in place
  float* h1_s  = h1_d + (size_t)ND * HD;          // agg then h1_s in place
  float* m3    = buf0;                            // layer-3 messages (NS*OD)
  float* agg3  = buf0 + (size_t)NS * OD;          // layer-3 agg (fits in buf0's 2nd half)

  float* out_h3s = (float*)d_out;
  float* out_h3d = out_h3s + (size_t)NS * OD;

  // degrees -> inv sqrt (inv_s, inv_d contiguous)
  gcn_zero_f32<<<cdiv(NS + ND, 256), 256, 0, stream>>>(ws, NS + ND);
  gcn_degree<<<cdiv(E, 256), 256, 0, stream>>>(src, dst, inv_s, inv_d, E);
  gcn_inv_sqrt<<<cdiv(NS + ND, 256), 256, 0, stream>>>(ws, NS + ND);

  dim3 g1(cdiv(NS, TM), HD / TN);
  dim3 g3(cdiv(NS, TM), OD / TN);

  // Layer 1: h1_d = relu(inv_d * scatter_dst((h_s*inv_s)@W1_sd) + b1_sd)
  gcn_gemm_rowscale<<<g1, 256, 0, stream>>>(h_s, inv_s, W1_sd, buf0, NS, CD, HD);
  gcn_zero_f32<<<cdiv(ND * HD, 256), 256, 0, stream>>>(h1_d, ND * HD);
  gcn_scatter_add<<<cdiv(E, 256 / (HD / 4)), 256, 0, stream>>>(buf0, src, dst, h1_d, E, HD);
  gcn_epilogue<<<cdiv(ND * HD, 256), 256, 0, stream>>>(h1_d, inv_d, b1_sd, h1_d, ND, HD);

  // Layer 1 reverse: h1_s = relu(inv_s * scatter_src((h_d*inv_d)@W1_ds) + b1_ds)
  gcn_gemm_rowscale<<<g1, 256, 0, stream>>>(h_d, inv_d, W1_ds, buf0, ND, CD, HD);
  gcn_zero_f32<<<cdiv(NS * HD, 256), 256, 0, stream>>>(h1_s, NS * HD);
  gcn_scatter_add<<<cdiv(E, 256 / (HD / 4)), 256, 0, stream>>>(buf0, dst, src, h1_s, E, HD);
  gcn_epilogue<<<cdiv(NS * HD, 256), 256, 0, stream>>>(h1_s, inv_s, b1_ds, h1_s, NS, HD);

  // Layer 3: h3_d = relu(inv_d * scatter_dst((h1_s*inv_s)@W3_sd) + b3_sd)
  gcn_gemm_rowscale<<<g3, 256, 0, stream>>>(h1_s, inv_s, W3_sd, m3, NS, HD, OD);
  gcn_zero_f32<<<cdiv(ND * OD, 256), 256, 0, stream>>>(agg3, ND * OD);
  gcn_scatter_add<<<cdiv(E, 256 / (OD / 4)), 256, 0, stream>>>(m3, src, dst, agg3, E, OD);
  gcn_epilogue<<<cdiv(ND * OD, 256), 256, 0, stream>>>(agg3, inv_d, b3_sd, out_h3d, ND, OD);

  // Layer 3 reverse: h3_s = relu(inv_s * scatter_src((h1_d*inv_d)@W3_ds) + b3_ds)
  gcn_gemm_rowscale<<<g3, 256, 0, stream>>>(h1_d, inv_d, W3_ds, m3, ND, HD, OD);
  gcn_zero_f32<<<cdiv(NS * OD, 256), 256, 0, stream>>>(agg3, NS * OD);
  gcn_scatter_add<<<cdiv(E, 256 / (OD / 4)), 256, 0, stream>>>(m3, dst, src, agg3, E, OD);
  gcn_epilogue<<<cdiv(NS * OD, 256), 256, 0, stream>>>(agg3, inv_s, b3_ds, out_h3s, NS, OD);
}